// TAPFormer_71141838291204
// MI455X (gfx1250) — compile-verified
//
#include <hip/hip_runtime.h>
#include <hip/hip_bf16.h>
#include <math.h>

typedef __attribute__((ext_vector_type(2))) float v2f;
typedef __attribute__((ext_vector_type(8))) float v8f;
typedef int v4i __attribute__((vector_size(16)));

#define SS   16
#define NPT  256
#define DD   128
#define RR   49          // 7*7
#define K4   2401        // 49*49
#define KPAD 2416        // 151*16
#define N1   384
#define N2   256
#define CH   854

// ---- optional CDNA5 async global->LDS copy path (ASYNCcnt-tracked) --------
#if defined(__has_builtin)
#if __has_builtin(__builtin_amdgcn_global_load_async_to_lds_b128)
#define USE_ASYNC_LDS 1
#endif
#endif
#ifndef USE_ASYNC_LDS
#define USE_ASYNC_LDS 0
#endif

#if USE_ASYNC_LDS
#define GLOBAL_AS __attribute__((address_space(1)))
#define LDS_AS    __attribute__((address_space(3)))
__device__ __forceinline__ void async_copy16(const float* g, float* l) {
    __builtin_amdgcn_global_load_async_to_lds_b128(
        (GLOBAL_AS v4i*)g, (LDS_AS v4i*)l, 0, 0);
}
__device__ __forceinline__ void async_wait0() {
    asm volatile("s_wait_asynccnt 0" ::: "memory");
}
#endif

__device__ __forceinline__ int iclamp(int v, int lo, int hi) {
    return v < lo ? lo : (v > hi ? hi : v);
}

// ---------------------------------------------------------------------------
// Time embedding: temb[s][c], c<427 -> sin(s*omega_c), else cos(s*omega_{c-427})
// ---------------------------------------------------------------------------
__global__ void temb_kernel(float* __restrict__ temb) {
    int i = blockIdx.x * blockDim.x + threadIdx.x;
    if (i >= SS * CH) return;
    int s = i / CH, c = i % CH;
    int k = (c < 427) ? c : c - 427;
    float omega = powf(10000.0f, -(float)k / 427.0f);
    float ang = (float)s * omega;
    temb[i] = (c < 427) ? sinf(ang) : cosf(ang);
}

// ---------------------------------------------------------------------------
// Per (s,n): bilinear-gather 49x128 features, load 49x128 template,
// compute vol[49x49] = feat @ tf^T with V_WMMA_F32_16X16X4_F32 (padded 64x64).
// ---------------------------------------------------------------------------
__global__ __launch_bounds__(128)
void gather_corr_kernel(const float* __restrict__ fmap,    // [S, D, H, W]
                        const float* __restrict__ coords,  // [S, NPT, 2]
                        const float* __restrict__ tfs,     // [49, NPT, D]
                        float* __restrict__ vol,           // [S*NPT, KPAD]
                        int H, int W, float inv_scale) {
    __shared__ float featLDS[64 * 68];   // [m=hw pt][k=chan], stride 68
    __shared__ float tfLDS[64 * 65];     // [k=chan][n=ij pt], stride 65

    const int blk  = blockIdx.x;          // s*NPT + n
    const int s    = blk >> 8;
    const int npt  = blk & 255;
    const int tid  = threadIdx.x;
    const int lane = tid & 31;
    const int wave = tid >> 5;
    const int l15  = lane & 15;
    const int koff = (lane >> 4) * 2;

    const float cx = coords[(s * NPT + npt) * 2 + 0] * inv_scale;
    const float cy = coords[(s * NPT + npt) * 2 + 1] * inv_scale;
    const int HW = H * W;

    v8f acc[4];
    for (int t = 0; t < 4; ++t)
        for (int e = 0; e < 8; ++e) acc[t][e] = 0.0f;

    for (int kh = 0; kh < 2; ++kh) {
        const int cbase = kh * 64;
        // gather features: 49 points x 64 channels (this half)
        for (int idx = tid; idx < RR * 64; idx += 128) {
            int c = idx & 63;
            int p = idx >> 6;                 // 0..48, p = h*7+w (h <-> y)
            int dy = p / 7 - 3;
            int dx = p % 7 - 3;
            float x = cx + (float)dx;
            float y = cy + (float)dy;
            float x0f = floorf(x), y0f = floorf(y);
            float wx = x - x0f, wy = y - y0f;
            int x0 = (int)x0f, y0 = (int)y0f;
            int x1 = x0 + 1, y1 = y0 + 1;
            bool vx0 = (x0 >= 0) && (x0 < W);
            bool vx1 = (x1 >= 0) && (x1 < W);
            bool vy0 = (y0 >= 0) && (y0 < H);
            bool vy1 = (y1 >= 0) && (y1 < H);
            int cx0 = iclamp(x0, 0, W - 1), cx1 = iclamp(x1, 0, W - 1);
            int cy0 = iclamp(y0, 0, H - 1), cy1 = iclamp(y1, 0, H - 1);
            const float* fmc = fmap + ((size_t)s * DD + (cbase + c)) * HW;
            float v00 = (vy0 && vx0) ? fmc[cy0 * W + cx0] : 0.0f;
            float v01 = (vy0 && vx1) ? fmc[cy0 * W + cx1] : 0.0f;
            float v10 = (vy1 && vx0) ? fmc[cy1 * W + cx0] : 0.0f;
            float v11 = (vy1 && vx1) ? fmc[cy1 * W + cx1] : 0.0f;
            float val = v00 * (1.0f - wx) * (1.0f - wy) + v01 * wx * (1.0f - wy)
                      + v10 * (1.0f - wx) * wy          + v11 * wx * wy;
            featLDS[p * 68 + c] = val;
        }
        for (int idx = tid; idx < 15 * 64; idx += 128) {   // zero pad rows 49..63
            int c = idx & 63, p = 49 + (idx >> 6);
            featLDS[p * 68 + c] = 0.0f;
        }
        // template: tfs[ij][npt][c] (coalesced over c)
        for (int idx = tid; idx < RR * 64; idx += 128) {
            int c = idx & 63, ij = idx >> 6;
            tfLDS[c * 65 + ij] = tfs[((size_t)ij * NPT + npt) * DD + cbase + c];
        }
        for (int idx = tid; idx < 15 * 64; idx += 128) {
            int c = idx & 63, ij = 49 + (idx >> 6);
            tfLDS[c * 65 + ij] = 0.0f;
        }
        __syncthreads();

        // 4 waves: wave w owns M-tile row (m0 = w*16) x 4 N-tiles
        const int mrow = wave * 16 + l15;
        for (int k = 0; k < 64; k += 4) {
            v2f a;
            a.x = featLDS[mrow * 68 + k + koff];
            a.y = featLDS[mrow * 68 + k + koff + 1];
            for (int nt = 0; nt < 4; ++nt) {
                v2f b;
                b.x = tfLDS[(k + koff) * 65 + nt * 16 + l15];
                b.y = tfLDS[(k + koff + 1) * 65 + nt * 16 + l15];
                acc[nt] = __builtin_amdgcn_wmma_f32_16x16x4_f32(
                    false, a, false, b, (short)0, acc[nt], false, false);
            }
        }
        __syncthreads();
    }

    // stage full 64x64 tile in LDS (featLDS is dead now), then store the
    // 2401 valid elements with one uniform, coalesced loop (no exec juggling)
    float* tile = featLDS;               // reuse as [64][65]
    for (int nt = 0; nt < 4; ++nt)
        for (int v = 0; v < 8; ++v) {
            int m = wave * 16 + v + 8 * (lane >> 4);
            int n = nt * 16 + l15;
            tile[m * 65 + n] = acc[nt][v];
        }
    __syncthreads();
    float* volrow = vol + (size_t)blk * KPAD;
    for (int idx = tid; idx < K4; idx += 128) {
        int m = idx / 49;
        int n = idx - m * 49;
        volrow[idx] = tile[m * 65 + n];
    }
    for (int t = K4 + tid; t < KPAD; t += 128) volrow[t] = 0.0f;
}

// ---------------------------------------------------------------------------
// h = gelu(vol @ W1 + b1).  M=4096, K=KPAD (guard k<2401), N=384.
// 64x64 block tile, 4 waves, each wave: 16x64 strip (4 accumulators).
// Tile staging uses GLOBAL_LOAD_ASYNC_TO_LDS_B128 when available.
// ---------------------------------------------------------------------------
__global__ __launch_bounds__(128)
void mlp1_kernel(const float* __restrict__ A,     // [4096, KPAD]
                 const float* __restrict__ W1,    // [2401, 384]
                 const float* __restrict__ b1,    // [384]
                 float* __restrict__ Hout) {      // [4096, 384]
    __shared__ float As[64 * 20];   // stride 20 floats (16B-aligned rows)
    __shared__ float Bs[16 * 68];   // stride 68 floats (16B-aligned rows)
    const int nblk = blockIdx.x;          // 0..5
    const int mblk = blockIdx.y;          // 0..63
    const int tid  = threadIdx.x;
    const int lane = tid & 31;
    const int wave = tid >> 5;
    const int l15  = lane & 15;
    const int koff = (lane >> 4) * 2;

    const int am = tid >> 1, akk = (tid & 1) * 8;      // A-tile role
    const int bk = tid >> 3, bnn = (tid & 7) * 8;      // B-tile role
    const float* Abase = A + (size_t)(mblk * 64 + am) * KPAD + akk;
    const int ng = nblk * 64 + bnn;

    v8f acc[4];
    for (int t = 0; t < 4; ++t)
        for (int e = 0; e < 8; ++e) acc[t][e] = 0.0f;

    for (int k0 = 0; k0 < KPAD; k0 += 16) {
        const float* asrc = Abase + k0;
        float* adst = &As[am * 20 + akk];
        const int kg = k0 + bk;
        const float* bsrc = W1 + (size_t)kg * N1 + ng;
        float* bdst = &Bs[bk * 68 + bnn];
#if USE_ASYNC_LDS
        async_copy16(asrc, adst);
        async_copy16(asrc + 4, adst + 4);
        if (kg < K4) {
            async_copy16(bsrc, bdst);
            async_copy16(bsrc + 4, bdst + 4);
        } else {
            for (int i = 0; i < 8; ++i) bdst[i] = 0.0f;
        }
        async_wait0();
#else
        for (int i = 0; i < 8; ++i) adst[i] = asrc[i];
        if (kg < K4) {
            for (int i = 0; i < 8; ++i) bdst[i] = bsrc[i];
        } else {
            for (int i = 0; i < 8; ++i) bdst[i] = 0.0f;
        }
#endif
        if (k0 + 16 < KPAD) {                 // prefetch next K tile
            __builtin_prefetch(asrc + 16, 0, 0);
            if (kg + 16 < K4) __builtin_prefetch(bsrc + (size_t)16 * N1, 0, 0);
        }
        __syncthreads();
        const int mrow = wave * 16 + l15;
        for (int ks = 0; ks < 16; ks += 4) {
            v2f a;
            a.x = As[mrow * 20 + ks + koff];
            a.y = As[mrow * 20 + ks + koff + 1];
            for (int nt = 0; nt < 4; ++nt) {
                v2f b;
                b.x = Bs[(ks + koff) * 68 + nt * 16 + l15];
                b.y = Bs[(ks + koff + 1) * 68 + nt * 16 + l15];
                acc[nt] = __builtin_amdgcn_wmma_f32_16x16x4_f32(
                    false, a, false, b, (short)0, acc[nt], false, false);
            }
        }
        __syncthreads();
    }

    for (int nt = 0; nt < 4; ++nt)
        for (int v = 0; v < 8; ++v) {
            int m = mblk * 64 + wave * 16 + v + 8 * (lane >> 4);
            int n = nblk * 64 + nt * 16 + l15;
            float x = acc[nt][v] + b1[n];
            float g = 0.5f * x * (1.0f + erff(x * 0.7071067811865476f));
            Hout[(size_t)m * N1 + n] = g;
        }
}

// ---------------------------------------------------------------------------
// out[:, :, ch] = h @ W2 + b2 + temb.  M=4096, K=384, N=256.
// Scatter to out[npt][s][2 + L*256 + n].
// ---------------------------------------------------------------------------
__global__ __launch_bounds__(128)
void mlp2_kernel(const float* __restrict__ Hin,   // [4096, 384]
                 const float* __restrict__ W2,    // [384, 256]
                 const float* __restrict__ b2,    // [256]
                 const float* __restrict__ temb,  // [16, 854]
                 float* __restrict__ out,         // [256, 16, 854]
                 int chbase) {
    __shared__ float As[64 * 20];
    __shared__ float Bs[16 * 68];
    const int nblk = blockIdx.x;          // 0..3
    const int mblk = blockIdx.y;          // 0..63
    const int tid  = threadIdx.x;
    const int lane = tid & 31;
    const int wave = tid >> 5;
    const int l15  = lane & 15;
    const int koff = (lane >> 4) * 2;

    const int am = tid >> 1, akk = (tid & 1) * 8;
    const int bk = tid >> 3, bnn = (tid & 7) * 8;
    const float* Abase = Hin + (size_t)(mblk * 64 + am) * N1 + akk;
    const float* Bbase = W2 + (size_t)bk * N2 + nblk * 64 + bnn;

    v8f acc[4];
    for (int t = 0; t < 4; ++t)
        for (int e = 0; e < 8; ++e) acc[t][e] = 0.0f;

    for (int k0 = 0; k0 < N1; k0 += 16) {
        const float* asrc = Abase + k0;
        const float* bsrc = Bbase + (size_t)k0 * N2;
        float* adst = &As[am * 20 + akk];
        float* bdst = &Bs[bk * 68 + bnn];
#if USE_ASYNC_LDS
        async_copy16(asrc, adst);
        async_copy16(asrc + 4, adst + 4);
        async_copy16(bsrc, bdst);
        async_copy16(bsrc + 4, bdst + 4);
        async_wait0();
#else
        for (int i = 0; i < 8; ++i) adst[i] = asrc[i];
        for (int i = 0; i < 8; ++i) bdst[i] = bsrc[i];
#endif
        __syncthreads();
        const int mrow = wave * 16 + l15;
        for (int ks = 0; ks < 16; ks += 4) {
            v2f a;
            a.x = As[mrow * 20 + ks + koff];
            a.y = As[mrow * 20 + ks + koff + 1];
            for (int nt = 0; nt < 4; ++nt) {
                v2f b;
                b.x = Bs[(ks + koff) * 68 + nt * 16 + l15];
                b.y = Bs[(ks + koff + 1) * 68 + nt * 16 + l15];
                acc[nt] = __builtin_amdgcn_wmma_f32_16x16x4_f32(
                    false, a, false, b, (short)0, acc[nt], false, false);
            }
        }
        __syncthreads();
    }

    for (int nt = 0; nt < 4; ++nt)
        for (int v = 0; v < 8; ++v) {
            int m = mblk * 64 + wave * 16 + v + 8 * (lane >> 4);
            int n = nblk * 64 + nt * 16 + l15;
            int s = m >> 8, npt = m & 255;
            int ch = chbase + n;
            float x = acc[nt][v] + b2[n] + temb[s * CH + ch];
            out[((size_t)npt * SS + s) * CH + ch] = x;
        }
}

// ---------------------------------------------------------------------------
// vis/conf + 84-channel posenc of relative flow + time embedding.
// ---------------------------------------------------------------------------
__global__ void pe_kernel(const float* __restrict__ coords,  // [16,256,2]
                          const float* __restrict__ vis,     // [16,256,1]
                          const float* __restrict__ conf,    // [16,256,1]
                          const float* __restrict__ temb,    // [16,854]
                          float* __restrict__ out) {         // [256,16,854]
    int i = blockIdx.x * blockDim.x + threadIdx.x;
    if (i >= SS * NPT) return;
    int s = i >> 8, npt = i & 255;
    float c0x = coords[(s * NPT + npt) * 2 + 0];
    float c0y = coords[(s * NPT + npt) * 2 + 1];
    float rfx = 0.0f, rfy = 0.0f, rbx = 0.0f, rby = 0.0f;
    if (s < SS - 1) {
        rfx = c0x - coords[((s + 1) * NPT + npt) * 2 + 0];
        rfy = c0y - coords[((s + 1) * NPT + npt) * 2 + 1];
    }
    if (s > 0) {
        rbx = c0x - coords[((s - 1) * NPT + npt) * 2 + 0];
        rby = c0y - coords[((s - 1) * NPT + npt) * 2 + 1];
    }
    float x4[4] = { rfx / 64.0f, rfy / 48.0f, rbx / 64.0f, rby / 48.0f };
    float pe[84];
    for (int j = 0; j < 4; ++j) pe[j] = x4[j];
    for (int d = 0; d < 10; ++d)
        for (int j = 0; j < 4; ++j) {
            float v = x4[j] * (float)(1 << d);
            pe[4 + d * 4 + j]  = sinf(v);
            pe[44 + d * 4 + j] = cosf(v);
        }
    size_t ob = ((size_t)npt * SS + s) * CH;
    out[ob + 0] = vis[s * NPT + npt] + temb[s * CH + 0];
    out[ob + 1] = conf[s * NPT + npt] + temb[s * CH + 1];
    for (int t = 0; t < 84; ++t)
        out[ob + 770 + t] = pe[t] + temb[s * CH + 770 + t];
}

// ---------------------------------------------------------------------------
extern "C" void kernel_launch(void* const* d_in, const int* in_sizes, int n_in,
                              void* d_out, int out_size, void* d_ws, size_t ws_size,
                              hipStream_t stream) {
    const float* fmaps[3] = { (const float*)d_in[0], (const float*)d_in[1], (const float*)d_in[2] };
    const float* coords   = (const float*)d_in[3];
    const float* tfs[3]   = { (const float*)d_in[4], (const float*)d_in[5], (const float*)d_in[6] };
    const float* vis      = (const float*)d_in[7];
    const float* conf     = (const float*)d_in[8];
    const float* W1       = (const float*)d_in[9];
    const float* b1       = (const float*)d_in[10];
    const float* W2       = (const float*)d_in[11];
    const float* b2       = (const float*)d_in[12];
    float* out = (float*)d_out;

    float* vol  = (float*)d_ws;                        // 4096 * 2416
    float* h    = vol + (size_t)SS * NPT * KPAD;       // 4096 * 384
    float* temb = h + (size_t)SS * NPT * N1;           // 16 * 854

    temb_kernel<<<(SS * CH + 255) / 256, 256, 0, stream>>>(temb);

    const int Hs[3] = { 48, 24, 12 };
    const int Ws[3] = { 64, 32, 16 };
    for (int L = 0; L < 3; ++L) {
        gather_corr_kernel<<<SS * NPT, 128, 0, stream>>>(
            fmaps[L], coords, tfs[L], vol, Hs[L], Ws[L], 1.0f / (float)(1 << L));
        mlp1_kernel<<<dim3(N1 / 64, SS * NPT / 64), 128, 0, stream>>>(vol, W1, b1, h);
        mlp2_kernel<<<dim3(N2 / 64, SS * NPT / 64), 128, 0, stream>>>(
            h, W2, b2, temb, out, 2 + L * 256);
    }
    pe_kernel<<<(SS * NPT + 255) / 256, 256, 0, stream>>>(coords, vis, conf, temb, out);
}